// Encoder_23141283790924
// MI455X (gfx1250) — compile-verified
//
#include <hip/hip_runtime.h>

#define BB     16
#define SCHAR  512
#define SWORD  768
#define DIM    768
#define KDIM   300
#define NT64   (DIM / 64)            // 12 n-tiles of 64
#define MBLK   ((BB * SCHAR) / 128)  // 64 m-blocks of 128 rows
#define KC     12                    // k-chunk rows staged in LDS
#define NCHUNK (KDIM / KC)           // 25 chunks

typedef __attribute__((ext_vector_type(2))) float v2f;
typedef __attribute__((ext_vector_type(8))) float v8f;

// ---------------------------------------------------------------------------
// Kernel 1: per-batch category + stable-sort destination (closed form ranks),
// wave32 ballot-based prefix counts.
// ---------------------------------------------------------------------------
__global__ void cat_scan_kernel(const int* __restrict__ word_idx,
                                const int* __restrict__ char_len,
                                int* __restrict__ cat_out,
                                int* __restrict__ dest_out) {
    int b    = blockIdx.x;
    int lane = threadIdx.x;                 // 0..31 (wave32)
    int cl   = char_len[b];
    const int* wi_row = word_idx + b * SWORD;

    int nWordTotal = 0;
    for (int base = 0; base < SWORD; base += 32) {
        int s  = base + lane;
        int wi = wi_row[s];
        bool isb = (s < cl);
        bool isw = (!isb) && (s < SCHAR) && (wi != 0);
        nWordTotal += __popcll(__ballot(isw));
    }

    int run = 0;
    for (int base = 0; base < SWORD; base += 32) {
        int s  = base + lane;
        int wi = wi_row[s];
        bool isb = (s < cl);
        bool isw = (!isb) && (s < SCHAR) && (wi != 0);
        int  cat = isb ? 0 : (isw ? 1 : 2);
        unsigned long long bal = __ballot(isw);
        int pre = __popcll(bal & ((1ull << lane) - 1ull));
        int nWordBefore = run + pre;

        int dest;
        if (cat == 0)      dest = s;
        else if (cat == 1) dest = cl + nWordBefore;
        else               dest = cl + nWordTotal + (s - cl - nWordBefore);

        cat_out[b * SWORD + s]  = cat;
        dest_out[b * SWORD + s] = dest;
        run += __popcll(bal);
    }
}

// ---------------------------------------------------------------------------
// Kernel 2: fp32 WMMA GEMM with async global->LDS double-buffered B staging.
// Block = 256 threads = 8 waves; block tile = 128(M) x 64(N); all waves share
// the B strip, which is copied by global_load_async_to_lds_b128 (ASYNCcnt).
// ---------------------------------------------------------------------------
__global__ __launch_bounds__(256) void wmma_gemm_kernel(
        const float* __restrict__ table,
        const float* __restrict__ W,
        const float* __restrict__ bias,
        const int*   __restrict__ word_idx,
        float*       __restrict__ wv) {
    __shared__ float smB[2][KC * 64];   // 2 x 3KB double buffer

    int tid    = threadIdx.x;
    int lane   = tid & 31;
    int wib    = tid >> 5;              // wave in block: 0..7
    int nT     = blockIdx.x % NT64;
    int mBlock = blockIdx.x / NT64;
    int n0     = nT * 64;
    int m0     = mBlock * 128 + wib * 16;

    int half = lane >> 4;               // A/B K-pairing per ISA layout
    int mr   = lane & 15;

    // Gathered A row for this lane (lanes 16-31 mirror 0-15).
    int m = m0 + mr;
    int b = m >> 9;
    int s = m & 511;
    const float* arow = table + (long)word_idx[b * SWORD + s] * KDIM;

    // Copier mapping: waves 0..5 each move one 512B segment (2 rows of 64).
    bool copier  = (wib < 6);
    int segFloat = wib * 128 + lane * 4;        // float offset within chunk strip
    int rowInC   = segFloat >> 6;               // row within KC-chunk
    int colInC   = segFloat & 63;

    // Prologue: issue async copy of chunk 0 into buffer 0.
    if (copier) {
        const float* gsrc = W + (long)rowInC * DIM + n0 + colInC;
        unsigned ldsOff = (unsigned)(size_t)
            (__attribute__((address_space(3))) float*)&smB[0][segFloat];
        asm volatile("global_load_async_to_lds_b128 %0, %1, off"
                     :: "v"(ldsOff), "v"(gsrc) : "memory");
    }

    v8f acc0 = {}, acc1 = {}, acc2 = {}, acc3 = {};

    for (int c = 0; c < NCHUNK; ++c) {
        asm volatile("s_wait_asynccnt 0x0" ::: "memory");
        __syncthreads();   // publish chunk c; also fences reads of the buffer
                           // about to be refilled (read in the previous iteration)
        if (copier && (c + 1 < NCHUNK)) {
            const float* gsrc = W + (long)((c + 1) * KC + rowInC) * DIM + n0 + colInC;
            unsigned ldsOff = (unsigned)(size_t)
                (__attribute__((address_space(3))) float*)&smB[(c + 1) & 1][segFloat];
            asm volatile("global_load_async_to_lds_b128 %0, %1, off"
                         :: "v"(ldsOff), "v"(gsrc) : "memory");
        }

        const float* bufB = smB[c & 1];
        const float* aK   = arow + c * KC;
        #pragma unroll
        for (int kk = 0; kk < KC; kk += 4) {
            int ka = kk + 2 * half;             // A: V0={K=0|K=2}, V1={K=1|K=3}
            v2f a;
            a.x = aK[ka];
            a.y = aK[ka + 1];
            const float* l0 = bufB + ka * 64 + mr;
            const float* l1 = l0 + 64;
            v2f b0; b0.x = l0[0];  b0.y = l1[0];
            v2f b1; b1.x = l0[16]; b1.y = l1[16];
            v2f b2; b2.x = l0[32]; b2.y = l1[32];
            v2f b3; b3.x = l0[48]; b3.y = l1[48];
            acc0 = __builtin_amdgcn_wmma_f32_16x16x4_f32(false, a, false, b0, (short)0, acc0, false, false);
            acc1 = __builtin_amdgcn_wmma_f32_16x16x4_f32(false, a, false, b1, (short)0, acc1, false, false);
            acc2 = __builtin_amdgcn_wmma_f32_16x16x4_f32(false, a, false, b2, (short)0, acc2, false, false);
            acc3 = __builtin_amdgcn_wmma_f32_16x16x4_f32(false, a, false, b3, (short)0, acc3, false, false);
        }
    }

    // Epilogue: C/D layout — VGPR r: lanes 0-15 -> M=r, lanes 16-31 -> M=r+8.
    int nc = lane & 15;
    float g0 = bias[n0 + nc],      g1 = bias[n0 + 16 + nc];
    float g2 = bias[n0 + 32 + nc], g3 = bias[n0 + 48 + nc];
    #pragma unroll
    for (int r = 0; r < 8; ++r) {
        int row = m0 + r + 8 * half;
        float* orow = wv + (long)row * DIM + n0 + nc;
        orow[0]  = acc0[r] + g0;
        orow[16] = acc1[r] + g1;
        orow[32] = acc2[r] + g2;
        orow[48] = acc3[r] + g3;
    }
}

// ---------------------------------------------------------------------------
// Kernel 3: LayerNorm + permuted pack + aux outputs.
// ---------------------------------------------------------------------------
__device__ __forceinline__ float block_reduce_sum(float v, float* sm) {
    for (int o = 16; o > 0; o >>= 1) v += __shfl_xor(v, o, 32);
    int lane = threadIdx.x & 31;
    int wid  = threadIdx.x >> 5;
    if (lane == 0) sm[wid] = v;
    __syncthreads();
    if (wid == 0) {
        float t = (lane < 8) ? sm[lane] : 0.0f;
        for (int o = 4; o > 0; o >>= 1) t += __shfl_xor(t, o, 32);
        if (lane == 0) sm[0] = t;
    }
    __syncthreads();
    float r = sm[0];
    __syncthreads();
    return r;
}

__global__ void ln_pack_kernel(const float* __restrict__ char_vec,
                               const float* __restrict__ wv,
                               const float* __restrict__ gamma,
                               const float* __restrict__ beta,
                               const int*   __restrict__ cat,
                               const int*   __restrict__ dest,
                               const int*   __restrict__ mask,
                               const int*   __restrict__ word_mask,
                               const int*   __restrict__ wpb,
                               const int*   __restrict__ wpe,
                               float*       __restrict__ out) {
    __shared__ float sm[8];
    int bs  = blockIdx.x;
    int b   = bs / SWORD;
    int s   = bs % SWORD;
    int tid = threadIdx.x;

    int c   = cat[bs];
    int dst = dest[bs];
    float* orow = out + ((long)b * SWORD + dst) * DIM;

    if (tid == 0) {
        const long AUX = (long)BB * SWORD * DIM;
        int cm  = (s < SCHAR) ? mask[b * SCHAR + s] : 0;
        int cwm = ((cm != 0) || (word_mask[bs] != 0)) ? 1 : 0;
        int pos = (s < SCHAR) ? s * cm : 0;
        out[AUX + bs]                    = (float)cwm;
        out[AUX + (long)BB * SWORD + bs] = (float)(pos + wpb[bs]);
        out[AUX + 2L * BB * SWORD + bs]  = (float)(pos + wpe[bs]);
    }

    if (c == 2) {  // uniform per block
        for (int i = tid; i < DIM; i += 256) orow[i] = 0.0f;
        return;
    }

    const float* src = (c == 0) ? (char_vec + ((long)b * SCHAR + s) * DIM)
                                : (wv       + ((long)b * SCHAR + s) * DIM);
    float x0 = src[tid], x1 = src[tid + 256], x2 = src[tid + 512];

    float mu = block_reduce_sum(x0 + x1 + x2, sm) * (1.0f / DIM);
    float d0 = x0 - mu, d1 = x1 - mu, d2 = x2 - mu;
    float var = block_reduce_sum(d0 * d0 + d1 * d1 + d2 * d2, sm) * (1.0f / DIM);
    float inv = rsqrtf(var + 1e-12f);

    orow[tid]       = d0 * inv * gamma[tid]       + beta[tid];
    orow[tid + 256] = d1 * inv * gamma[tid + 256] + beta[tid + 256];
    orow[tid + 512] = d2 * inv * gamma[tid + 512] + beta[tid + 512];
}

// ---------------------------------------------------------------------------
extern "C" void kernel_launch(void* const* d_in, const int* in_sizes, int n_in,
                              void* d_out, int out_size, void* d_ws, size_t ws_size,
                              hipStream_t stream) {
    const float* char_vec   = (const float*)d_in[0];
    const float* w2v_table  = (const float*)d_in[1];
    const float* w2v_W      = (const float*)d_in[2];
    const float* w2v_b      = (const float*)d_in[3];
    const float* ln_gamma   = (const float*)d_in[4];
    const float* ln_beta    = (const float*)d_in[5];
    const int*   mask       = (const int*)d_in[6];
    const int*   char_len   = (const int*)d_in[7];
    const int*   word_idx   = (const int*)d_in[8];
    const int*   word_mask  = (const int*)d_in[9];
    const int*   word_pos_b = (const int*)d_in[10];
    const int*   word_pos_e = (const int*)d_in[11];
    float* out = (float*)d_out;

    // Workspace layout: cat[B*Sw] | dest[B*Sw] | word_vec[B*512*768]  (~25.3 MB)
    int*   catp  = (int*)d_ws;
    int*   destp = catp + BB * SWORD;
    float* wvp   = (float*)(destp + BB * SWORD);

    cat_scan_kernel<<<BB, 32, 0, stream>>>(word_idx, char_len, catp, destp);

    // 64 m-blocks(128 rows) * 12 n-tiles(64) = 768 blocks of 8 waves
    wmma_gemm_kernel<<<MBLK * NT64, 256, 0, stream>>>(
        w2v_table, w2v_W, w2v_b, word_idx, wvp);

    ln_pack_kernel<<<BB * SWORD, 256, 0, stream>>>(
        char_vec, wvp, ln_gamma, ln_beta, catp, destp,
        mask, word_mask, word_pos_b, word_pos_e, out);
}